// SparseGlass_87058987090241
// MI455X (gfx1250) — compile-verified
//
#include <hip/hip_runtime.h>

#define NS 512   // samples
#define GS 256   // genes
#define P2 512   // 2*G
#define STR 36   // LDS row stride (floats): 144B, 16B-aligned, bank-conflict-free

#if defined(__AMDGCN__) && __has_builtin(__builtin_amdgcn_global_load_async_to_lds_b128) && __has_builtin(__builtin_amdgcn_s_wait_asynccnt)
#define HAVE_ASYNC 1
#else
#define HAVE_ASYNC 0
#endif

typedef __attribute__((ext_vector_type(2))) float v2f;
typedef __attribute__((ext_vector_type(8))) float v8f;
typedef __attribute__((ext_vector_type(4))) int   v4i;

__device__ __forceinline__ float rcpf(float x)  { return __builtin_amdgcn_rcpf(x);  }
__device__ __forceinline__ float lg2f(float x)  { return __builtin_amdgcn_logf(x);  }
__device__ __forceinline__ float ex2f(float x)  { return __builtin_amdgcn_exp2f(x); }

// ---------------------------------------------------------------------------
// Stage ROWS x 32 floats from global (row stride GS) into LDS (row stride STR)
// via gfx1250 async global->LDS DMA (ASYNCcnt), 16B per lane per instruction.
// ROWS*8 is a multiple of 256, so every wave issues the same instruction count
// (copy_tile<32> -> 1 async op/wave, copy_tile<64> -> 2 async ops/wave).
// ---------------------------------------------------------------------------
template<int ROWS>
__device__ __forceinline__ void copy_tile(const float* __restrict__ g0,
                                          float* l0, int row0, int gc, int tid)
{
    for (int s = tid; s < ROWS * 8; s += 256) {
        const int r = s >> 3, c = (s & 7) << 2;
        const float* gp = g0 + (row0 + r) * GS + gc + c;
        float* lp = l0 + r * STR + c;
#if HAVE_ASYNC
        __builtin_amdgcn_global_load_async_to_lds_b128(
            (__attribute__((address_space(1))) v4i*)(gp),
            (__attribute__((address_space(3))) v4i*)(lp), 0, 0);
#else
        lp[0] = gp[0]; lp[1] = gp[1]; lp[2] = gp[2]; lp[3] = gp[3];
#endif
    }
}

// ---------------------------------------------------------------------------
// Kernel 1: per-(i,g) precursors + per-row log-product  (one block per row i)
// ---------------------------------------------------------------------------
__global__ __launch_bounds__(GS)
void prep_kernel(const float* __restrict__ value,
                 const float* __restrict__ gscale,
                 const float* __restrict__ gmean,
                 const float* __restrict__ glass_h,
                 float* __restrict__ Sg,    // [N][2G]  sigma flattened (p = 2a+b)
                 float* __restrict__ ds1,   // [N][G]   dsigma magnitude
                 float* __restrict__ tone,  // [N][G]   term_one
                 float* __restrict__ ixl,   // [N][G]   1/(x-l)
                 float* __restrict__ lP2)   // [N]      sum_g log2(x-l)
{
    __shared__ float red[GS];
    const int i = blockIdx.x, g = threadIdx.x;
    const float x  = value[i * GS + g];
    const float gs = gscale[0], gm = gmean[0];
    const float z  = gs * (x - gm);
    const float s  = rcpf(1.0f + ex2f(-z * 1.4426950408889634f)); // sigmoid
    const float d1 = gs * (1.0f - s) * s;
    Sg[i * P2 + 2 * g]     = 1.0f - s;
    Sg[i * P2 + 2 * g + 1] = s;
    ds1[i * GS + g]  = d1;
    tone[i * GS + g] = d1 * (glass_h[2 * g + 1] - glass_h[2 * g]); // dsigma.(h0,h1)
    const float xl = x + 0.5f;           // x - l, l = -0.5
    ixl[i * GS + g] = rcpf(xl);
    red[g] = lg2f(xl);
    __syncthreads();
    for (int off = GS / 2; off > 0; off >>= 1) {
        if (g < off) red[g] += red[g + off];
        __syncthreads();
    }
    if (g == 0) lP2[i] = red[0];
}

// ---------------------------------------------------------------------------
// Kernel 2: build B[p][q] = Jmat[j,a,k,b], q=2j+k, p=2a+b (K-major for GEMM)
// ---------------------------------------------------------------------------
__global__ void build_B(const float* __restrict__ Jraw, float* __restrict__ B)
{
    const int idx = blockIdx.x * blockDim.x + threadIdx.x; // 0 .. 2G*2G-1
    const int p = idx >> 9, q = idx & 511;
    const int a = p >> 1, b = p & 1, j = q >> 1, k = q & 1;
    float v = 0.0f;
    if (j < a) {
        const int pos = j * GS - (j * (j + 1)) / 2 + (a - j - 1);
        v = Jraw[pos * 4 + k * 2 + b];          // Jraw[pos][k][b]
    } else if (j > a) {
        const int pos = a * GS - (a * (a + 1)) / 2 + (j - a - 1);
        v = Jraw[pos * 4 + b * 2 + k];          // transpose(Jraw[pos])[k][b]
    }
    B[idx] = v;
}

// ---------------------------------------------------------------------------
// Kernel 3: T = Sg @ B  (512x512x512 f32 GEMM via V_WMMA_F32_16X16X4_F32)
// block = 8 waves, macro tile 32(M) x 64(N), wave tile 16x16
// ---------------------------------------------------------------------------
__global__ __launch_bounds__(256)
void gemm_wmma(const float* __restrict__ A, const float* __restrict__ Bm,
               float* __restrict__ C)
{
    __shared__ float As[32][33];
    __shared__ float Bs[32][65];
    const int tid = threadIdx.x;
    const int wave = tid >> 5, lane = tid & 31;
    const int i0 = blockIdx.x * 32, j0 = blockIdx.y * 64;
    const int tm = (wave >> 2) * 16, tn = (wave & 3) * 16;
    const int la = lane & 15, kh = (lane >> 4) * 2, mh = (lane >> 4) * 8;
    v8f acc = {};
    for (int kc = 0; kc < P2; kc += 32) {
        for (int t = tid; t < 32 * 32; t += 256) {
            const int r = t >> 5, c = t & 31;
            As[r][c] = A[(i0 + r) * P2 + kc + c];
        }
        for (int t = tid; t < 32 * 64; t += 256) {
            const int r = t >> 6, c = t & 63;
            Bs[r][c] = Bm[(kc + r) * P2 + j0 + c];
        }
        __syncthreads();
        #pragma unroll
        for (int ks = 0; ks < 32; ks += 4) {
            v2f af, bf;
            af.x = As[tm + la][ks + kh];     af.y = As[tm + la][ks + kh + 1];
            bf.x = Bs[ks + kh][tn + la];     bf.y = Bs[ks + kh + 1][tn + la];
            acc = __builtin_amdgcn_wmma_f32_16x16x4_f32(
                      false, af, false, bf, (short)0, acc, false, false);
        }
        __syncthreads();
    }
    #pragma unroll
    for (int r = 0; r < 8; ++r)
        C[(i0 + tm + mh + r) * P2 + j0 + tn + la] = acc[r];
}

// ---------------------------------------------------------------------------
// Kernel 3b: sscore = term_one + dsigma . t
// ---------------------------------------------------------------------------
__global__ void score_kernel(const float* __restrict__ T,
                             const float* __restrict__ tone,
                             const float* __restrict__ ds1,
                             float* __restrict__ sscore)
{
    const int idx = blockIdx.x * blockDim.x + threadIdx.x; // N*G
    const int i = idx >> 8, j = idx & 255;
    sscore[idx] = tone[idx] + ds1[idx] * (T[i * P2 + 2 * j + 1] - T[i * P2 + 2 * j]);
}

// ---------------------------------------------------------------------------
// Kernel 4: fused pairwise NKSD. Block = 8 waves over a 32x64 (i,j) tile.
// - Double-buffered LDS tiles streamed via GLOBAL_LOAD_ASYNC_TO_LDS_B128
//   (9 async ops/wave/chunk; in-order ASYNCcnt lets us wait <=9 to retire the
//   previous chunk while the next is in flight).
// - Per wave: 16x16 pair tile. WMMA accumulates S = ss.ss^T, U = ixl.ixl^T,
//   V = ixl_i.ss_j^T over g; scalar loop accumulates non-separable sums.
// - denom in [1,2): product of 32 denoms fits fp32 -> ONE v_log_f32 per
//   32 elements instead of one per element (TRANS ops are quarter-rate).
// - Lane ownership of scalar pairs matches the WMMA C layout exactly.
// ---------------------------------------------------------------------------
__global__ __launch_bounds__(256)
void pair_kernel(const float* __restrict__ value,
                 const float* __restrict__ sscore,
                 const float* __restrict__ ixl,
                 const float* __restrict__ lP2,
                 float* __restrict__ partials)
{
    __shared__ float ldsI[2][3][32 * STR];   // [buf][0=value,1=sscore,2=ixl]
    __shared__ float ldsJ[2][3][64 * STR];
    __shared__ float redc[256], redk[256];

    const float beta = -0.5f / 256.0f;
    const float cA1  = -2.0f * beta;
    const float cA2  = -4.0f * beta * (beta - 1.0f);
    const float twob =  2.0f * beta;

    const int tid = threadIdx.x;
    const int wave = tid >> 5, lane = tid & 31;
    const int i0 = blockIdx.x * 32, j0 = blockIdx.y * 64;
    const int ti = (wave >> 2) * 16, tj = (wave & 3) * 16;
    const int la = lane & 15, kh = (lane >> 4) * 2, mh = (lane >> 4) * 8;

    v8f Cs = {}, Cu = {}, Cv = {};
    float slog[8] = {}, A1[8] = {}, A2[8] = {}, T2[8] = {}, KY[8] = {}, KX[8] = {};

    // prologue: stage chunk 0 into buffer 0
    copy_tile<32>(value,  &ldsI[0][0][0], i0, 0, tid);
    copy_tile<32>(sscore, &ldsI[0][1][0], i0, 0, tid);
    copy_tile<32>(ixl,    &ldsI[0][2][0], i0, 0, tid);
    copy_tile<64>(value,  &ldsJ[0][0][0], j0, 0, tid);
    copy_tile<64>(sscore, &ldsJ[0][1][0], j0, 0, tid);
    copy_tile<64>(ixl,    &ldsJ[0][2][0], j0, 0, tid);

    for (int c = 0; c < 8; ++c) {
        const int cur = c & 1;
        if (c < 7) {                       // prefetch next chunk into other buf
            const int nxt = cur ^ 1, gc = (c + 1) * 32;
            copy_tile<32>(value,  &ldsI[nxt][0][0], i0, gc, tid);
            copy_tile<32>(sscore, &ldsI[nxt][1][0], i0, gc, tid);
            copy_tile<32>(ixl,    &ldsI[nxt][2][0], i0, gc, tid);
            copy_tile<64>(value,  &ldsJ[nxt][0][0], j0, gc, tid);
            copy_tile<64>(sscore, &ldsJ[nxt][1][0], j0, gc, tid);
            copy_tile<64>(ixl,    &ldsJ[nxt][2][0], j0, gc, tid);
        }
#if HAVE_ASYNC
        if (c < 7) __builtin_amdgcn_s_wait_asynccnt(9);  // current chunk landed
        else       __builtin_amdgcn_s_wait_asynccnt(0);
#endif
        __syncthreads();

        const float* vi = &ldsI[cur][0][0];
        const float* si = &ldsI[cur][1][0];
        const float* xi = &ldsI[cur][2][0];
        const float* vj = &ldsJ[cur][0][0];
        const float* sj = &ldsJ[cur][1][0];
        const float* xj = &ldsJ[cur][2][0];

        // --- three fused rank-32 WMMA updates ---
        #pragma unroll
        for (int ks = 0; ks < 32; ks += 4) {
            v2f aS, aX, bS, bX;
            aS.x = si[(ti + la) * STR + ks + kh]; aS.y = si[(ti + la) * STR + ks + kh + 1];
            aX.x = xi[(ti + la) * STR + ks + kh]; aX.y = xi[(ti + la) * STR + ks + kh + 1];
            bS.x = sj[(tj + la) * STR + ks + kh]; bS.y = sj[(tj + la) * STR + ks + kh + 1];
            bX.x = xj[(tj + la) * STR + ks + kh]; bX.y = xj[(tj + la) * STR + ks + kh + 1];
            Cs = __builtin_amdgcn_wmma_f32_16x16x4_f32(false, aS, false, bS, (short)0, Cs, false, false);
            Cu = __builtin_amdgcn_wmma_f32_16x16x4_f32(false, aX, false, bX, (short)0, Cu, false, false);
            Cv = __builtin_amdgcn_wmma_f32_16x16x4_f32(false, aX, false, bS, (short)0, Cv, false, false);
        }

        // --- non-separable per-(pair,g) sums: now rcp-bound only ---
        float pd[8];
        #pragma unroll
        for (int r = 0; r < 8; ++r) pd[r] = 1.0f;
        for (int g = 0; g < 32; ++g) {
            const float vjg = vj[(tj + la) * STR + g];
            const float sjg = sj[(tj + la) * STR + g];
            const float xjg = xj[(tj + la) * STR + g];
            #pragma unroll
            for (int r = 0; r < 8; ++r) {
                const float xv  = vi[(ti + mh + r) * STR + g];
                const float d   = xv - vjg;
                const float den = fmaf(d, d, 1.0f);   // c*c + (x-y)^2, c=1
                const float rr  = rcpf(den);
                pd[r] *= den;                          // log taken once per chunk
                A1[r] += rr;
                const float drr = d * rr;
                A2[r] = fmaf(drr, drr, A2[r]);
                T2[r] = fmaf(drr, sjg, T2[r]);
                KY[r] = fmaf(drr, xjg, KY[r]);
                KX[r] = fmaf(drr, xi[(ti + mh + r) * STR + g], KX[r]);
            }
        }
        #pragma unroll
        for (int r = 0; r < 8; ++r) slog[r] += lg2f(pd[r]);

        __syncthreads();   // all waves done reading buf[cur] before next issue
    }

    // --- epilogue: combine scalar accumulators with WMMA fragments ---
    const int jg = j0 + tj + la;
    const float lpj = lP2[jg];
    float contrib = 0.0f, klsum = 0.0f;
    #pragma unroll
    for (int r = 0; r < 8; ++r) {
        const int ig = i0 + ti + mh + r;
        if (ig == jg) continue;                           // offdiag mask
        const float Kv    = ex2f(beta * slog[r]);         // prod denom^beta
        const float lcorr = ex2f(lP2[ig] + lpj);          // separable l_corr
        const float Kl    = Kv * lcorr;
        const float Kpp   = (cA1 * A1[r] + cA2 * A2[r]) * Kv;
        const float t = Kl * Cs[r]
                      + 2.0f * (lcorr * (twob * Kv) * T2[r] + Kl * Cv[r])
                      + (Kpp + twob * Kv * (KY[r] - KX[r]) + Kv * Cu[r]) * lcorr;
        contrib += t;
        klsum   += Kl;
    }
    redc[tid] = contrib; redk[tid] = klsum;
    __syncthreads();
    for (int off = 128; off > 0; off >>= 1) {
        if (tid < off) { redc[tid] += redc[tid + off]; redk[tid] += redk[tid + off]; }
        __syncthreads();
    }
    if (tid == 0) {
        const int bid = blockIdx.y * gridDim.x + blockIdx.x;
        partials[2 * bid]     = redc[0];
        partials[2 * bid + 1] = redk[0];
    }
}

// ---------------------------------------------------------------------------
// Kernel 5: deterministic final reduction over 128 block partials
// ---------------------------------------------------------------------------
__global__ __launch_bounds__(128)
void finalize_kernel(const float* __restrict__ partials, float* __restrict__ out)
{
    __shared__ float rc[128], rk[128];
    const int t = threadIdx.x;
    rc[t] = partials[2 * t]; rk[t] = partials[2 * t + 1];
    __syncthreads();
    for (int off = 64; off > 0; off >>= 1) {
        if (t < off) { rc[t] += rc[t + off]; rk[t] += rk[t + off]; }
        __syncthreads();
    }
    if (t == 0) out[0] = -512.0f * rc[0] / rk[0];   // -(N/T) * ksd / Kbarl
}

// ---------------------------------------------------------------------------
extern "C" void kernel_launch(void* const* d_in, const int* in_sizes, int n_in,
                              void* d_out, int out_size, void* d_ws, size_t ws_size,
                              hipStream_t stream)
{
    const float* value   = (const float*)d_in[0];
    const float* gscale  = (const float*)d_in[1];
    const float* gmean   = (const float*)d_in[2];
    const float* glass_h = (const float*)d_in[3];
    const float* Jraw    = (const float*)d_in[4];
    float* out = (float*)d_out;

    float* ws    = (float*)d_ws;
    float* Sg    = ws;                 // 512*512
    float* Bm    = Sg   + 262144;      // 512*512
    float* Tm    = Bm   + 262144;      // 512*512
    float* ds1   = Tm   + 262144;      // 512*256
    float* tone  = ds1  + 131072;      // 512*256
    float* ixl   = tone + 131072;      // 512*256
    float* ssc   = ixl  + 131072;      // 512*256
    float* lP2   = ssc  + 131072;      // 512
    float* parts = lP2  + 512;         // 256

    prep_kernel<<<NS, GS, 0, stream>>>(value, gscale, gmean, glass_h,
                                       Sg, ds1, tone, ixl, lP2);
    build_B<<<(P2 * P2) / 256, 256, 0, stream>>>(Jraw, Bm);
    gemm_wmma<<<dim3(16, 8), 256, 0, stream>>>(Sg, Bm, Tm);
    score_kernel<<<(NS * GS) / 256, 256, 0, stream>>>(Tm, tone, ds1, ssc);
    pair_kernel<<<dim3(16, 8), 256, 0, stream>>>(value, ssc, ixl, lP2, parts);
    finalize_kernel<<<1, 128, 0, stream>>>(parts, out);
}